// GraphSAGE_56753697849938
// MI455X (gfx1250) — compile-verified
//
#include <hip/hip_runtime.h>

// ---------------------------------------------------------------------------
// GraphSAGE forward for MI455X (gfx1250, wave32).
//   scatter-mean (fp32 global atomics, L2-resident gathers)
//   SAGE layer GEMMs via V_WMMA_F32_16X16X4_F32 (f32-native matrix pipe)
// ---------------------------------------------------------------------------

typedef __attribute__((ext_vector_type(2))) float v2f;
typedef __attribute__((ext_vector_type(8))) float v8f;

#define NNODES 100000
#define NEDGES 1600000
#define INCH   64
#define HID    128
#define OUTCH  2

__device__ __forceinline__ void atomAddF(float* p, float v) {
  unsafeAtomicAdd(p, v);   // -> global_atomic_add_f32 (no CAS loop)
}

// ---------------------------------------------------------------------------
// Scatter: agg[dst][:] += feat[src][:]  (+ cnt[dst] += 1 when cnt != null)
// One thread per (edge, 16B chunk): b128 gather + 4 fp32 atomics.
// ---------------------------------------------------------------------------
template <int C>
__global__ __launch_bounds__(256)
void scatter_kernel(const float* __restrict__ feat,
                    const long long* __restrict__ src,
                    const long long* __restrict__ dst,
                    float* __restrict__ agg,
                    float* __restrict__ cnt) {
  constexpr int QPE = C / 4;  // float4 chunks per edge
  long long tid = (long long)blockIdx.x * blockDim.x + threadIdx.x;
  long long e  = tid / QPE;
  int       q  = (int)(tid % QPE);
  if (e >= NEDGES) return;
  int s = (int)src[e];
  int d = (int)dst[e];
  const float4 v = *(const float4*)(feat + (long long)s * C + q * 4);
  float* base = agg + (long long)d * C + q * 4;
  atomAddF(base + 0, v.x);
  atomAddF(base + 1, v.y);
  atomAddF(base + 2, v.z);
  atomAddF(base + 3, v.w);
  if (cnt != nullptr && q == 0) atomAddF(cnt + d, 1.0f);
}

// ---------------------------------------------------------------------------
// SAGE layer: hout = relu( (agg/cnt) @ Wn + hin @ Ws + bias )
// Block = 16 nodes x 128 outputs; 8 waves, each wave owns one 16x16 tile.
// K-loop: two chained v_wmma_f32_16x16x4_f32 per 4-wide K step.
// ---------------------------------------------------------------------------
template <int K>
__global__ __launch_bounds__(256)
void sage_gemm_kernel(const float* __restrict__ agg,
                      const float* __restrict__ cnt,
                      const float* __restrict__ hin,
                      const float* __restrict__ Wn,   // [K, HID] row-major
                      const float* __restrict__ Ws,   // [K, HID] row-major
                      const float* __restrict__ bias, // [HID]
                      float* __restrict__ hout) {     // [N, HID]
  __shared__ float sMean[16][K + 1];   // +1 pad: conflict-free column reads
  __shared__ float sH[16][K + 1];
  __shared__ float sInv[16];

  const int tid   = threadIdx.x;
  const int node0 = blockIdx.x * 16;

  if (tid < 16) {
    float c = cnt[node0 + tid];
    sInv[tid] = 1.0f / fmaxf(c, 1.0f);
  }
  __syncthreads();

  for (int i = tid; i < 16 * K; i += 256) {
    int r = i / K;
    int k = i % K;
    long long g = (long long)(node0 + r) * K + k;
    sMean[r][k] = agg[g] * sInv[r];
    sH[r][k]    = hin[g];
  }
  __syncthreads();

  const int lane = tid & 31;
  const int wv   = tid >> 5;           // 8 waves -> 8 column tiles of 16
  const int n0   = wv * 16;
  const int m    = lane & 15;          // A: M = lane%16
  const int khi  = (lane >> 4) * 2;    // A/B: K msb from lane group
  const int nl   = lane & 15;          // B/C/D: N = lane%16

  v8f acc = {0.f, 0.f, 0.f, 0.f, 0.f, 0.f, 0.f, 0.f};

#pragma unroll 4
  for (int kk = 0; kk < K; kk += 4) {
    const int kb = kk + khi;
    v2f aM, aH, bN, bS;
    aM[0] = sMean[m][kb];  aM[1] = sMean[m][kb + 1];
    aH[0] = sH[m][kb];     aH[1] = sH[m][kb + 1];
    bN[0] = Wn[(long long)kb * HID + n0 + nl];
    bN[1] = Wn[(long long)(kb + 1) * HID + n0 + nl];
    bS[0] = Ws[(long long)kb * HID + n0 + nl];
    bS[1] = Ws[(long long)(kb + 1) * HID + n0 + nl];
    // D = A*B + C, f32 16x16x4 (8 args: neg_a, A, neg_b, B, c_mod, C, reuse_a, reuse_b)
    acc = __builtin_amdgcn_wmma_f32_16x16x4_f32(false, aM, false, bN,
                                                (short)0, acc, false, false);
    acc = __builtin_amdgcn_wmma_f32_16x16x4_f32(false, aH, false, bS,
                                                (short)0, acc, false, false);
  }

  const float bv    = bias[n0 + nl];
  const int   rbase = (lane < 16) ? 0 : 8;   // C/D layout: VGPR r -> row r / r+8
#pragma unroll
  for (int r = 0; r < 8; r++) {
    float v = fmaxf(acc[r] + bv, 0.0f);
    hout[(long long)(node0 + rbase + r) * HID + n0 + nl] = v;
  }
}

// ---------------------------------------------------------------------------
// Final fc: out = h1 @ fc_w + fc_b   (128 -> 2), pure VALU, bandwidth trivial.
// ---------------------------------------------------------------------------
__global__ __launch_bounds__(256)
void fc_kernel(const float* __restrict__ h, const float* __restrict__ fcw,
               const float* __restrict__ fcb, float* __restrict__ out) {
  int i = blockIdx.x * blockDim.x + threadIdx.x;
  if (i >= NNODES) return;
  const float* hp = h + (long long)i * HID;
  float a0 = fcb[0], a1 = fcb[1];
#pragma unroll 8
  for (int k = 0; k < HID; k++) {
    float hv = hp[k];
    a0 += hv * fcw[k * 2 + 0];
    a1 += hv * fcw[k * 2 + 1];
  }
  out[(long long)i * 2 + 0] = a0;
  out[(long long)i * 2 + 1] = a1;
}

// ---------------------------------------------------------------------------
extern "C" void kernel_launch(void* const* d_in, const int* in_sizes, int n_in,
                              void* d_out, int out_size, void* d_ws, size_t ws_size,
                              hipStream_t stream) {
  (void)in_sizes; (void)n_in; (void)out_size; (void)ws_size;

  const float*     x    = (const float*)d_in[0];
  const long long* ei   = (const long long*)d_in[1];  // int64 per reference
  const float*     Wn0  = (const float*)d_in[2];
  const float*     Ws0  = (const float*)d_in[3];
  const float*     b0   = (const float*)d_in[4];
  const float*     Wn1  = (const float*)d_in[5];
  const float*     Ws1  = (const float*)d_in[6];
  const float*     b1   = (const float*)d_in[7];
  const float*     fcw  = (const float*)d_in[8];
  const float*     fcb  = (const float*)d_in[9];
  float*           out  = (float*)d_out;

  const long long* src = ei;            // edge_index[0]
  const long long* dst = ei + NEDGES;   // edge_index[1]

  // Workspace layout (floats): cnt | agg0 | h0 | agg1 | h1  == 449*N floats
  float* ws   = (float*)d_ws;
  float* cnt  = ws;
  float* agg0 = cnt  + NNODES;
  float* h0   = agg0 + (size_t)NNODES * INCH;
  float* agg1 = h0   + (size_t)NNODES * HID;
  float* h1   = agg1 + (size_t)NNODES * HID;

  // Zero accumulators (cnt+agg0 contiguous; agg1 separate). Capture-safe.
  hipMemsetAsync(cnt,  0, sizeof(float) * (size_t)NNODES * (1 + INCH), stream);
  hipMemsetAsync(agg1, 0, sizeof(float) * (size_t)NNODES * HID, stream);

  // Layer 0
  {
    long long t = (long long)NEDGES * (INCH / 4);
    scatter_kernel<INCH><<<(unsigned)((t + 255) / 256), 256, 0, stream>>>(
        x, src, dst, agg0, cnt);
  }
  sage_gemm_kernel<INCH><<<NNODES / 16, 256, 0, stream>>>(
      agg0, cnt, x, Wn0, Ws0, b0, h0);

  // Layer 1
  {
    long long t = (long long)NEDGES * (HID / 4);
    scatter_kernel<HID><<<(unsigned)((t + 255) / 256), 256, 0, stream>>>(
        h0, src, dst, agg1, nullptr);
  }
  sage_gemm_kernel<HID><<<NNODES / 16, 256, 0, stream>>>(
      agg1, cnt, h0, Wn1, Ws1, b1, h1);

  // Output head
  fc_kernel<<<(NNODES + 255) / 256, 256, 0, stream>>>(h1, fcw, fcb, out);
}